// EarthMoverDistance_37056977829909
// MI455X (gfx1250) — compile-verified
//
#include <hip/hip_runtime.h>
#include <math.h>

// ---------------------------------------------------------------------------
// Entropic EMD (Sinkhorn, log-domain) on MI455X / gfx1250.
//
// arg_ij = (pot_j - C_ij)/eps,  C_ij = |p_i|^2 + |q_j|^2 - 2 p_i.q_j
// One V_WMMA_F32_16X16X4_F32 per 16x16 tile:
//   A row i  = (2px/e, 2py/e, 2pz/e, 1)
//   B col j  = (qx, qy, qz, (pot_j - |q_j|^2)/e)     (pre-swizzled in LDS)
//   C init   = -|p_i|^2/e
//   D        = arg_ij
// fp32 WMMA is mandatory: 1/eps = 200 amplifies cost errors before exp.
// Two-pass LSE (exact row max, then sum of exp) keeps the inner loops free
// of cross-lane ops; shuffles happen once per kernel, not once per tile.
// ---------------------------------------------------------------------------

typedef __attribute__((ext_vector_type(2))) float v2f;
typedef __attribute__((ext_vector_type(8))) float v8f;

#define EPS       0.005f
#define INV_EPS   200.0f
#define OT_ITERS  50
#define BATCH     8
#define NPTS      2048
#define TILE      16
#define WAVES_PB  8   // 256 threads = 8 wave32
#define NTILES    (NPTS / TILE)

// workspace layout in floats (BN = BATCH*NPTS = 16384)
#define BN        (BATCH * NPTS)
#define WS_F      0              // f potentials  [BN]
#define WS_G      (BN)           // g potentials  [BN]
#define WS_PA4    (2  * BN)      // preds scaled: (2x/e,2y/e,2z/e,-|p|^2/e)
#define WS_QA4    (6  * BN)      // gts scaled
#define WS_PR4    (10 * BN)      // preds raw: (x,y,z,|p|^2)
#define WS_QR4    (14 * BN)      // gts raw   (total 18*BN floats ~ 1.2MB)

__global__ __launch_bounds__(256)
void emd_precompute(const float* __restrict__ preds,
                    const float* __restrict__ gts,
                    float* __restrict__ ws,
                    float* __restrict__ out) {
  int t = blockIdx.x * blockDim.x + threadIdx.x;
  if (t == 0) out[0] = 0.0f;           // re-zero accumulator every call
  if (t >= BN) return;
  float px = preds[3 * t + 0], py = preds[3 * t + 1], pz = preds[3 * t + 2];
  float qx = gts[3 * t + 0],   qy = gts[3 * t + 1],   qz = gts[3 * t + 2];
  float pn = px * px + py * py + pz * pz;
  float qn = qx * qx + qy * qy + qz * qz;
  ws[WS_F + t] = 0.0f;
  ws[WS_G + t] = 0.0f;
  float4* pa4 = (float4*)(ws + WS_PA4);
  float4* qa4 = (float4*)(ws + WS_QA4);
  float4* pr4 = (float4*)(ws + WS_PR4);
  float4* qr4 = (float4*)(ws + WS_QR4);
  pa4[t] = make_float4(2.f * px * INV_EPS, 2.f * py * INV_EPS, 2.f * pz * INV_EPS, -pn * INV_EPS);
  qa4[t] = make_float4(2.f * qx * INV_EPS, 2.f * qy * INV_EPS, 2.f * qz * INV_EPS, -qn * INV_EPS);
  pr4[t] = make_float4(px, py, pz, pn);
  qr4[t] = make_float4(qx, qy, qz, qn);
}

// One half-iteration: outPot_i = -eps * LSE_j( (colPot_j - C_ij)/eps )
// Block = 8 waves sharing one batch; each wave owns one 16-row tile.
__global__ __launch_bounds__(256)
void emd_update(float* __restrict__ outPot,
                const float4* __restrict__ rowA4,   // (2x/e,2y/e,2z/e,-n/e)
                const float4* __restrict__ colR4,   // (x,y,z,n)
                const float* __restrict__ colPot) {
  // Pre-swizzled B fragments: entry [jt*32 + lane] is that lane's 2 K-values.
  __shared__ float2 sB[NPTS * 2];   // 32 KB

  const int tid  = threadIdx.x;
  const int lane = tid & 31;
  const int wave = tid >> 5;
  const int half = lane >> 4;
  const int l16  = lane & 15;
  const int base = blockIdx.y * NPTS;
  const int m0   = (blockIdx.x * WAVES_PB + wave) * TILE;

  // Stage all column fragments for this batch (shared by the 8 waves).
  for (int j = tid; j < NPTS; j += 256) {
    float4 q = colR4[base + j];
    float w = (colPot[base + j] - q.w) * INV_EPS;
    int jt = j >> 4, jl = j & 15;
    sB[jt * 32 + jl]      = make_float2(q.x, q.y);  // lanes 0-15 : K=0,1
    sB[jt * 32 + 16 + jl] = make_float2(q.z, w);    // lanes 16-31: K=2,3
  }
  __syncthreads();

  // A fragment: lane -> row M=l16; VGPR0 = K(half?2:0), VGPR1 = K(half?3:1)
  float4 pr = rowA4[base + m0 + l16];
  v2f a;
  a.x = half ? pr.z : pr.x;
  a.y = half ? 1.0f : pr.y;

  // accumulator init: VGPR k holds row M = k + 8*half
  v8f cinit;
#pragma unroll
  for (int k = 0; k < 8; ++k)
    cinit[k] = rowA4[base + m0 + k + 8 * half].w;

  // ---- pass 1: lane-local row maxima (no cross-lane ops in the loop) ----
  float mx[8];
#pragma unroll
  for (int k = 0; k < 8; ++k) mx[k] = -3.0e38f;

  for (int jt = 0; jt < NTILES; ++jt) {
    float2 bv = sB[jt * 32 + lane];
    v2f bb; bb.x = bv.x; bb.y = bv.y;
    v8f d = __builtin_amdgcn_wmma_f32_16x16x4_f32(
        false, a, false, bb, (short)0, cinit, false, false);
#pragma unroll
    for (int k = 0; k < 8; ++k) mx[k] = fmaxf(mx[k], d[k]);
  }
  // exact row max: reduce across the 16 lanes of this half (once per kernel)
#pragma unroll
  for (int k = 0; k < 8; ++k) {
    mx[k] = fmaxf(mx[k], __shfl_xor(mx[k], 1, 32));
    mx[k] = fmaxf(mx[k], __shfl_xor(mx[k], 2, 32));
    mx[k] = fmaxf(mx[k], __shfl_xor(mx[k], 4, 32));
    mx[k] = fmaxf(mx[k], __shfl_xor(mx[k], 8, 32));
  }

  // ---- pass 2: lane-local sums of exp(arg - rowmax) ----
  float s[8];
#pragma unroll
  for (int k = 0; k < 8; ++k) s[k] = 0.0f;

  for (int jt = 0; jt < NTILES; ++jt) {
    float2 bv = sB[jt * 32 + lane];
    v2f bb; bb.x = bv.x; bb.y = bv.y;
    v8f d = __builtin_amdgcn_wmma_f32_16x16x4_f32(
        false, a, false, bb, (short)0, cinit, false, false);
#pragma unroll
    for (int k = 0; k < 8; ++k) s[k] += __expf(d[k] - mx[k]);
  }
#pragma unroll
  for (int k = 0; k < 8; ++k) {
    s[k] += __shfl_xor(s[k], 1, 32);
    s[k] += __shfl_xor(s[k], 2, 32);
    s[k] += __shfl_xor(s[k], 4, 32);
    s[k] += __shfl_xor(s[k], 8, 32);
  }

  if (l16 == 0) {
#pragma unroll
    for (int k = 0; k < 8; ++k)
      outPot[base + m0 + k + 8 * half] = -EPS * (mx[k] + __logf(s[k]));
  }
}

// total += sum exp((f_i + g_j - C)/eps) * C, with C = f_i + g_j - eps*arg
__global__ __launch_bounds__(256)
void emd_final(const float4* __restrict__ pa4,
               const float4* __restrict__ qr4,
               const float* __restrict__ f,
               const float* __restrict__ g,
               float* __restrict__ out) {
  __shared__ float2 sB[NPTS * 2];   // 32 KB fragments
  __shared__ float  sG[NPTS];       //  8 KB raw g (to recover C elementwise)

  const int tid  = threadIdx.x;
  const int lane = tid & 31;
  const int wave = tid >> 5;
  const int half = lane >> 4;
  const int l16  = lane & 15;
  const int base = blockIdx.y * NPTS;
  const int m0   = (blockIdx.x * WAVES_PB + wave) * TILE;

  for (int j = tid; j < NPTS; j += 256) {
    float4 q = qr4[base + j];
    float gj = g[base + j];
    float w = (gj - q.w) * INV_EPS;
    int jt = j >> 4, jl = j & 15;
    sB[jt * 32 + jl]      = make_float2(q.x, q.y);
    sB[jt * 32 + 16 + jl] = make_float2(q.z, w);
    sG[j] = gj;
  }
  __syncthreads();

  float4 pr = pa4[base + m0 + l16];
  v2f a;
  a.x = half ? pr.z : pr.x;
  a.y = half ? 1.0f : pr.y;

  float fi[8];
  v8f cinit;
#pragma unroll
  for (int k = 0; k < 8; ++k) {
    int r = base + m0 + k + 8 * half;
    float fv = f[r];
    fi[k] = fv;
    cinit[k] = fv * INV_EPS + pa4[r].w;   // (f_i - |p_i|^2)/eps
  }

  float local = 0.0f;
  for (int jt = 0; jt < NTILES; ++jt) {
    float2 bv = sB[jt * 32 + lane];
    float  gj = sG[jt * 16 + l16];
    v2f bb; bb.x = bv.x; bb.y = bv.y;
    v8f d = __builtin_amdgcn_wmma_f32_16x16x4_f32(
        false, a, false, bb, (short)0, cinit, false, false);
#pragma unroll
    for (int k = 0; k < 8; ++k) {
      float arg = d[k];                     // (f_i + g_j - C)/eps
      float C   = fi[k] + gj - EPS * arg;   // recover C
      local += __expf(arg) * C;
    }
  }
  local += __shfl_xor(local, 1, 32);
  local += __shfl_xor(local, 2, 32);
  local += __shfl_xor(local, 4, 32);
  local += __shfl_xor(local, 8, 32);
  local += __shfl_xor(local, 16, 32);
  if (lane == 0) atomicAdd(out, local);
}

extern "C" void kernel_launch(void* const* d_in, const int* in_sizes, int n_in,
                              void* d_out, int out_size, void* d_ws, size_t ws_size,
                              hipStream_t stream) {
  const float* preds = (const float*)d_in[0];
  const float* gts   = (const float*)d_in[1];
  float* out = (float*)d_out;
  float* ws  = (float*)d_ws;

  float*  fpot = ws + WS_F;
  float*  gpot = ws + WS_G;
  float4* pa4  = (float4*)(ws + WS_PA4);
  float4* qa4  = (float4*)(ws + WS_QA4);
  float4* pr4  = (float4*)(ws + WS_PR4);
  float4* qr4  = (float4*)(ws + WS_QR4);

  emd_precompute<<<(BN + 255) / 256, 256, 0, stream>>>(preds, gts, ws, out);

  dim3 grid(NPTS / (TILE * WAVES_PB), BATCH);  // (16, 8) blocks of 8 waves
  for (int it = 0; it < OT_ITERS; ++it) {
    // f = -eps*LSE_j((g_j - C_ij)/eps)
    emd_update<<<grid, 256, 0, stream>>>(fpot, pa4, qr4, gpot);
    // g = -eps*LSE_i((f_i - C_ij)/eps)   (roles swapped; C structurally symmetric)
    emd_update<<<grid, 256, 0, stream>>>(gpot, qa4, pr4, fpot);
  }
  emd_final<<<grid, 256, 0, stream>>>(pa4, qr4, fpot, gpot, out);
}